// RpnLoss_1649267441864
// MI455X (gfx1250) — compile-verified
//
#include <hip/hip_runtime.h>
#include <hip/hip_bf16.h>

typedef float v2f __attribute__((ext_vector_type(2)));
typedef float v8f __attribute__((ext_vector_type(8)));

// Full-wave (wave32) sum using V_WMMA_F32_16X16X4_F32.
// A is 16x4 f32: lanes 0-15 supply K=0 (a.x) / K=1 (a.y); lanes 16-31 supply K=2/K=3.
// With a.y = 0 and B = ones, D[m,n] = partial[m] + partial[m+16] for every column n.
// Lane L holds D[0..7, col] (L<16) or D[8..15, col] (L>=16) in its 8 result VGPRs,
// so (sum of own 8) + (same from lane^16) == full wave sum, on every lane.
// Requires EXEC all-ones: call only from uniform control flow.
__device__ __forceinline__ float wave_sum_wmma(float v) {
    v2f a; a.x = v;    a.y = 0.0f;
    v2f b; b.x = 1.0f; b.y = 1.0f;
    v8f c = {};
    v8f d = __builtin_amdgcn_wmma_f32_16x16x4_f32(
        /*neg_a=*/false, a, /*neg_b=*/false, b,
        /*c_mod=*/(short)0, c, /*reuse_a=*/false, /*reuse_b=*/false);
    float s = d[0] + d[1] + d[2] + d[3] + d[4] + d[5] + d[6] + d[7];
    s += __shfl_xor(s, 16, 32);
    return s;
}

__device__ __forceinline__ float smooth_l1(float d) {
    float ad = fabsf(d);
    return ad < 1.0f ? 0.5f * d * d : ad - 0.5f;
}

#define MAXL 512

// Kernel 1: scan all anchors; for positives, accumulate classification +
// 4*regression loss over all B batches; emit per-block partial loss and count.
__global__ void k1_anchor_scan(const float* __restrict__ cla,
                               const float* __restrict__ reg,
                               const float* __restrict__ label,
                               const float* __restrict__ anchor,
                               const int* __restrict__ pA,
                               const int* __restrict__ pF,
                               const int* __restrict__ pC,
                               int N, int B, int L,
                               float* __restrict__ blockLoss,
                               float* __restrict__ blockP) {
    __shared__ float s_cls[MAXL];
    __shared__ float s_occ[MAXL];
    __shared__ float wls[32], wcs[32];

    const int tid = threadIdx.x;
    const int n   = blockIdx.x * blockDim.x + tid;
    const int A   = *pA;
    const int F   = *pF;
    const float Cf = (float)(*pC);

    // Label classes -> LDS, then occ[l] = #{j<l : cls[j]==cls[l]}
    for (int l = tid; l < L; l += blockDim.x) s_cls[l] = label[l * 5];
    __syncthreads();
    for (int l = tid; l < L; l += blockDim.x) {
        float cl = s_cls[l];
        int cnt = 0;
        for (int j = 0; j < l; ++j) cnt += (s_cls[j] == cl) ? 1 : 0;
        s_occ[l] = (float)cnt;
    }
    __syncthreads();

    float acc  = 0.0f;   // this thread's pos-loss contribution (summed over B)
    float posf = 0.0f;   // 1 if positive anchor

    if (n < N) {
        const float* a = anchor + (long)n * 7;
        const float lbl  = a[1];
        const bool  pos  = (lbl != -1.0f) && (lbl != Cf);
        if (pos) {
            posf = 1.0f;
            const float bx0 = a[2], bx1 = a[3], bx2 = a[4], bx3 = a[5];
            const float aocc = a[6];

            // first matching label (argmax of boolean), default 0
            int gt = 0; bool found = false;
            for (int l = 0; l < L; ++l) {
                if (!found && s_cls[l] == lbl && s_occ[l] == aocc) { gt = l; found = true; }
            }
            const float* g = label + (long)gt * 5;
            const float t0 = (g[1] - bx0) / bx2;
            const float t1 = (g[2] - bx1) / bx3;
            const float t2 = logf(g[3] / bx2);
            const float t3 = logf(g[4] / bx3);

            const int ia = n % A;
            const int q  = n / A;
            const int h  = q % F;
            const int w  = q / F;
            const long plane = (long)F * F;
            const long hw    = (long)h * F + w;
            const long rbs   = (long)4 * A * plane;   // reg batch stride
            const long cbs   = (long)2 * A * plane;   // cla batch stride
            const float* rp = reg + (long)(ia * 4) * plane + hw;
            const float* cp = cla + (long)(ia * 2) * plane + hw;

            for (int b = 0; b < B; ++b) {
                const float p0 = rp[0 * plane];
                const float p1 = rp[1 * plane];
                const float p2 = rp[2 * plane];
                const float p3 = rp[3 * plane];
                const float q0 = cp[0];
                const float q1 = cp[plane];

                const float lr = 0.25f * (smooth_l1(p0 - t0) + smooth_l1(p1 - t1) +
                                          smooth_l1(p2 - t2) + smooth_l1(p3 - t3));
                const float m   = fmaxf(q0, q1);
                const float lse = m + logf(expf(q0 - m) + expf(q1 - m));
                acc += (lse - q1) + 4.0f * lr;

                rp += rbs;
                cp += cbs;
            }
        }
    }

    // Deterministic block reduction: WMMA per wave, then LDS across waves.
    const float ws = wave_sum_wmma(acc);
    const float wc = wave_sum_wmma(posf);
    const int wid  = tid >> 5;
    const int lane = tid & 31;
    if (lane == 0) { wls[wid] = ws; wcs[wid] = wc; }
    __syncthreads();
    if (tid == 0) {
        float S = 0.0f, Cn = 0.0f;
        const int nw = blockDim.x >> 5;
        for (int wv = 0; wv < nw; ++wv) { S += wls[wv]; Cn += wcs[wv]; }
        blockLoss[blockIdx.x] = S;
        blockP[blockIdx.x]    = Cn;
    }
}

// Kernel 2: single wave reduces the per-block partials -> totals[0]=pos_loss, totals[1]=P.
__global__ void k2_reduce(const float* __restrict__ blockLoss,
                          const float* __restrict__ blockP,
                          int nb, float* __restrict__ totals) {
    const int lane = threadIdx.x;
    float s = 0.0f, c = 0.0f;
    for (int i = lane; i < nb; i += 32) { s += blockLoss[i]; c += blockP[i]; }
    s = wave_sum_wmma(s);
    c = wave_sum_wmma(c);
    if (lane == 0) { totals[0] = s; totals[1] = c; }
}

// Kernel 3: negative-sample path (serial 300-sample cumsum per batch) + finalize.
__global__ void k3_neg_finalize(const float* __restrict__ cla,
                                const float* __restrict__ anchor,
                                const int* __restrict__ negs,
                                const int* __restrict__ pA,
                                const int* __restrict__ pF,
                                const int* __restrict__ pC,
                                int B, int NEG,
                                const float* __restrict__ totals,
                                float* __restrict__ out) {
    const int lane = threadIdx.x;
    const int A  = *pA;
    const int F  = *pF;
    const float Cf = (float)(*pC);
    const int Pi = (int)(totals[1] + 0.5f);
    const long plane = (long)F * F;
    const long cbs   = (long)2 * A * plane;

    float negsum = 0.0f;
    float negcnt = 0.0f;

    for (int b = lane; b < B; b += 32) {
        const float* cb = cla + (long)b * cbs;
        int cum = 0;
        float ls = 0.0f;
        int   lc = 0;
        for (int s = 0; s < NEG; ++s) {
            const int idx = negs[(long)b * NEG + s];
            const float lbl = anchor[(long)idx * 7 + 1];
            if (lbl == Cf) {
                ++cum;
                if (cum <= 3 * Pi) {
                    const int ia = idx % A;
                    const int q  = idx / A;
                    const int h  = q % F;
                    const int w  = q / F;
                    const long base = (long)(ia * 2) * plane + (long)h * F + w;
                    const float q0 = cb[base];
                    const float q1 = cb[base + plane];
                    const float m   = fmaxf(q0, q1);
                    const float lse = m + logf(expf(q0 - m) + expf(q1 - m));
                    ls += lse - q0;
                    ++lc;
                }
            }
        }
        negsum += ls;
        negcnt += (float)lc;
    }

    const float negS = wave_sum_wmma(negsum);
    const float negC = wave_sum_wmma(negcnt);
    if (lane == 0) {
        const float denom = (float)B * totals[1] + negC;
        out[0] = (totals[0] + negS) / denom;
    }
}

extern "C" void kernel_launch(void* const* d_in, const int* in_sizes, int n_in,
                              void* d_out, int out_size, void* d_ws, size_t ws_size,
                              hipStream_t stream) {
    const float* cla    = (const float*)d_in[0];
    const float* reg    = (const float*)d_in[1];
    const float* label  = (const float*)d_in[2];
    const float* anchor = (const float*)d_in[3];
    const int*   negs   = (const int*)d_in[4];
    const int*   pA     = (const int*)d_in[5];
    const int*   pF     = (const int*)d_in[6];
    const int*   pC     = (const int*)d_in[7];

    // Derive shapes from flat sizes (scalar params live on device):
    //   anchor_data: N*7 ; cla_map: B*2A*F*F ; label_data: L*5 ; neg_samples: B*NEG
    const long S0 = in_sizes[0];
    const long S2 = in_sizes[2];
    const long S3 = in_sizes[3];
    const long S4 = in_sizes[4];
    const int N   = (int)(S3 / 7);
    const int B   = (int)((7 * S0) / (2 * S3));
    const int L   = (int)(S2 / 5);
    const int NEG = (int)(S4 / B);

    const int threads = 256;
    const int nb1 = (N + threads - 1) / threads;

    float* fws       = (float*)d_ws;
    float* blockLoss = fws;            // nb1 floats
    float* blockP    = fws + nb1;      // nb1 floats
    float* totals    = fws + 2 * nb1;  // 2 floats: {pos_loss, P}

    k1_anchor_scan<<<nb1, threads, 0, stream>>>(cla, reg, label, anchor,
                                                pA, pF, pC, N, B, L,
                                                blockLoss, blockP);
    k2_reduce<<<1, 32, 0, stream>>>(blockLoss, blockP, nb1, totals);
    k3_neg_finalize<<<1, 32, 0, stream>>>(cla, anchor, negs, pA, pF, pC,
                                          B, NEG, totals, (float*)d_out);
}